// BallQuery_3315714753180
// MI455X (gfx1250) — compile-verified
//
#include <hip/hip_runtime.h>

typedef __attribute__((ext_vector_type(2))) float v2f;
typedef __attribute__((ext_vector_type(8))) float v8f;

#define RADIUS2 0.0225f
#define NS      32
#define NPTS    16384
#define NQ      1024
#define NB      8

// One wave32 handles 16 queries. Per 16-point tile:
//   A (16x4 f32): row m = (-2qx, -2qy, -2qz, |q|^2)   [lanes 0-15: K0,K1; lanes 16-31: K2,K3]
//   B (4x16 f32): col n = ( px,   py,   pz,   1   )   [same K striping, N over lanes]
//   D = A*B = -2 q.p + |q|^2 ; add |p|^2 per column -> d^2 (16x16 in 8 VGPRs)
// Then ordered selection via ballot + rank-popcount per C/D row.
__global__ __launch_bounds__(256) void BallQuery_kernel(
    const float* __restrict__ xyz,      // [B, N, 3]
    const float* __restrict__ new_xyz,  // [B, M, 3]
    int* __restrict__ out)              // [B, M, NS]
{
    __shared__ int sbuf[8 * 16 * NS];   // 8 waves/block * 16 queries * 32 slots = 16 KB

    const int lane      = threadIdx.x & 31;
    const int waveInBlk = threadIdx.x >> 5;
    const int w         = blockIdx.x * 8 + waveInBlk;   // 512 waves total
    const int b         = w >> 6;                        // 64 waves per batch
    const int qbase     = (w & 63) << 4;                 // 16 queries per wave
    int* buf = &sbuf[waveInBlk * 16 * NS];

    // ---- Load this wave's 16 queries (row = lane & 15) and build A operand ----
    const int   m  = qbase + (lane & 15);
    const float* qp = new_xyz + ((size_t)b * NQ + m) * 3;
    const float qx = qp[0], qy = qp[1], qz = qp[2];
    const float qss = qx * qx + qy * qy + qz * qz;
    v2f A;
    A.x = (lane < 16) ? (-2.0f * qx) : (-2.0f * qz);   // K0 / K2
    A.y = (lane < 16) ? (-2.0f * qy) : qss;            // K1 / K3

    int cnt = 0;  // valid in lanes 0-15: match count for query row == lane
    const float* xb = xyz + (size_t)b * NPTS * 3;

    for (int nbase = 0; nbase < NPTS; nbase += 16) {
        // ---- Load 16 points (col = lane & 15) and build B operand ----
        const int   col = nbase + (lane & 15);
        const float* pp = xb + (size_t)col * 3;
        const float px = pp[0], py = pp[1], pz = pp[2];
        const float pss = px * px + py * py + pz * pz;
        v2f Bm;
        Bm.x = (lane < 16) ? px : pz;                  // K0 / K2
        Bm.y = (lane < 16) ? py : 1.0f;                // K1 / K3

        if (nbase + 16 < NPTS)
            __builtin_prefetch(xb + (size_t)(col + 16) * 3, 0, 1);

        // ---- d^2 tile via f32 WMMA (EXEC is all ones here) ----
        v8f C = {};
        C = __builtin_amdgcn_wmma_f32_16x16x4_f32(
                /*neg_a=*/false, A, /*neg_b=*/false, Bm,
                /*c_mod=*/(short)0, C, /*reuse_a=*/false, /*reuse_b=*/false);

        // ---- Ordered selection: VGPR v holds rows v (lanes 0-15) and v+8 (16-31) ----
#pragma unroll
        for (int v = 0; v < 8; ++v) {
            const float d2   = C[v] + pss;
            const bool  pred = d2 < RADIUS2;
            const unsigned int mask = (unsigned int)__ballot(pred);
            const unsigned int low  = mask & 0xffffu;
            const unsigned int high = (mask >> 16) & 0xffffu;

            const int          row  = v + ((lane >= 16) ? 8 : 0);
            const unsigned int bits = (lane < 16) ? low : high;
            const int          pos  = lane & 15;
            const int rank   = __popc(bits & ((1u << pos) - 1u));
            const int cntRow = __shfl(cnt, row, 32);
            const int slot   = cntRow + rank;
            if (pred && slot < NS)
                buf[row * NS + slot] = nbase + pos;

            const int pcLow  = __popc(low);
            const int pcHigh = __popc(high);
            if (lane == v)     cnt += pcLow;
            if (lane == v + 8) cnt += pcHigh;
        }

        // Wave-uniform early exit: all 16 queries saturated.
        if (__all((lane >= 16) || (cnt >= NS)))
            break;
    }

    // ---- Emit: slot j -> buf[j] if j < cnt, else buf[0] (or N+1 if cnt == 0) ----
    int* ob = out + ((size_t)b * NQ + qbase) * NS;
#pragma unroll 1
    for (int r = 0; r < 16; ++r) {
        const int cntr = __shfl(cnt, r, 32);
        int val;
        if (lane < cntr)      val = buf[r * NS + lane];
        else if (cntr > 0)    val = buf[r * NS];
        else                  val = NPTS + 1;
        ob[r * NS + lane] = val;
    }
}

extern "C" void kernel_launch(void* const* d_in, const int* in_sizes, int n_in,
                              void* d_out, int out_size, void* d_ws, size_t ws_size,
                              hipStream_t stream) {
    const float* xyz     = (const float*)d_in[0];  // (8, 16384, 3) f32
    const float* new_xyz = (const float*)d_in[1];  // (8, 1024, 3)  f32
    int* out = (int*)d_out;                        // (8, 1024, 32) i32

    // 512 waves = 64 blocks x 256 threads (8 waves/block)
    BallQuery_kernel<<<64, 256, 0, stream>>>(xyz, new_xyz, out);
    (void)in_sizes; (void)n_in; (void)out_size; (void)d_ws; (void)ws_size;
}